// PrimitiveGate_32186484916956
// MI455X (gfx1250) — compile-verified
//
#include <hip/hip_runtime.h>

typedef __attribute__((ext_vector_type(2))) float v2f;
typedef __attribute__((ext_vector_type(8))) float v8f;

#define NQ        23
#define DIM       (1u << NQ)          // 8388608 rows
#define BATCH     4u
#define TQ        12
#define PLANE     (DIM * BATCH)       // 33554432 floats per (re|im) plane
#define HALFROWS  (1u << TQ)          // 4096 rows between pair partners
#define BLKF      (HALFROWS * BATCH)  // 16384 floats per half-block
#define NTILES    (PLANE / 2u / 16u)  // 1048576 WMMA tiles
#define TPH       (BLKF / 16u)        // 1024 tiles per h-block

// D = A(16x4) x B(4x16) + 0 using V_WMMA_F32_16X16X4_F32.
//   A rows 0-3  : M4       = [[R,-I],[I,R]]  (cols = [s0re,s1re,s0im,s1im])
//   A rows 8-11 : M4 rows permuted [2,3,0,1] (im rows first)
//   -> lanes 0-15 of D v0/v1 hold o0re/o1re, lanes 16-31 hold o0im/o1im.
__global__ __launch_bounds__(256) void gate_wmma_kernel(
    const float* __restrict__ sre, const float* __restrict__ sim,
    const float* __restrict__ mre, const float* __restrict__ mim,
    float* __restrict__ out)
{
    const int lane = (int)(threadIdx.x & 31u);

    // ---- gate matrix: uniform scalar loads (SGPR broadcast) ----
    const float r00 = mre[0], r01 = mre[1], r10 = mre[2], r11 = mre[3];
    const float i00 = mim[0], i01 = mim[1], i10 = mim[2], i11 = mim[3];

    // ---- build A operand (16x4 f32 -> 2 VGPRs) via lane-bit selects ----
    // VGPR0 holds K=0 (lanes 0-15) / K=2 (lanes 16-31); VGPR1 holds K=1/K=3.
    const int  r     = lane & 3;            // row within 4-row group
    const bool valid = ((lane >> 2) & 1) == 0;  // rows 4-7,12-15 are zero
    const int  p     = (lane >> 3) & 1;     // 1 -> permuted copy (rows 8-11)
    const int  cb    = (lane >> 4) & 1;     // column block (K0/1 vs K2/3)
    const int  rb    = ((r >> 1) ^ p) & 1;  // effective block-row of M4
    const int  ri    = r & 1;               // row within 2x2 block
    const float Rc0 = ri ? r10 : r00;       // R[ri][0]
    const float Rc1 = ri ? r11 : r01;       // R[ri][1]
    const float Ic0 = ri ? i10 : i00;       // I[ri][0]
    const float Ic1 = ri ? i11 : i01;       // I[ri][1]
    // block (rb,cb): diag -> R ; upper-right -> -I ; lower-left -> +I
    const float a0 = (rb == cb) ? Rc0 : (cb ? -Ic0 : Ic0);
    const float a1 = (rb == cb) ? Rc1 : (cb ? -Ic1 : Ic1);
    v2f A;
    A.x = valid ? a0 : 0.0f;
    A.y = valid ? a1 : 0.0f;

    // per-lane bases: lanes 0-15 handle the real part, 16-31 the imag part
    const float* __restrict__ bbase = (lane < 16) ? sre : sim;
    float*       __restrict__ obase = out + ((lane < 16) ? 0u : PLANE);
    const unsigned c = (unsigned)(lane & 15);

    const unsigned wave   = (blockIdx.x * blockDim.x + threadIdx.x) >> 5;
    const unsigned nwaves = (gridDim.x * blockDim.x) >> 5;

    for (unsigned t = wave; t < NTILES; t += nwaves) {
        const unsigned h     = t >> 10;        // t / TPH
        const unsigned inner = t & (TPH - 1u);
        const unsigned base0 = h * (2u * BLKF) + inner * 16u; // bit12 = 0 half
        const unsigned base1 = base0 + BLKF;                  // bit12 = 1 half

        // B operand (4x16 f32 -> 2 VGPRs): two full-wave b32 loads,
        // each hits two aligned 64B segments (re in lanes 0-15, im in 16-31).
        v2f B;
        B.x = bbase[base0 + c];   // K=0 (s0re) | K=2 (s0im)
        B.y = bbase[base1 + c];   // K=1 (s1re) | K=3 (s1im)

        v8f acc = {};
        acc = __builtin_amdgcn_wmma_f32_16x16x4_f32(
            /*neg_a=*/false, A, /*neg_b=*/false, B,
            /*c_mod=*/(short)0, acc, /*reuse_a=*/false, /*reuse_b=*/false);

        // D v0: o0re (lanes 0-15) / o0im (lanes 16-31)  -> row with bit12=0
        // D v1: o1re / o1im                              -> row with bit12=1
        obase[base0 + c] = acc[0];
        obase[base1 + c] = acc[1];
    }
}

extern "C" void kernel_launch(void* const* d_in, const int* in_sizes, int n_in,
                              void* d_out, int out_size, void* d_ws, size_t ws_size,
                              hipStream_t stream) {
    (void)in_sizes; (void)n_in; (void)out_size; (void)d_ws; (void)ws_size;
    const float* sre = (const float*)d_in[0];
    const float* sim = (const float*)d_in[1];
    const float* mre = (const float*)d_in[2];
    const float* mim = (const float*)d_in[3];
    float* out = (float*)d_out;

    // 4096 blocks x 256 threads = 32768 waves; 32 tiles/wave grid-stride.
    dim3 grid(4096), block(256);
    gate_wmma_kernel<<<grid, block, 0, stream>>>(sre, sim, mre, mim, out);
}